// IterativeTTTLayer_11570641895785
// MI455X (gfx1250) — compile-verified
//
#include <hip/hip_runtime.h>
#include <hip/hip_bf16.h>
#include <math.h>

// Problem constants (match reference)
#define BB     8
#define SS     256
#define HIDC   1024
#define HH     16
#define DD     64
#define FF     64
#define LR     0.1f
#define GSCALE (2.0f / 8192.0f)   // 2 / (B*H*D)
#define LN_EPS 1e-5f

typedef float v2f __attribute__((ext_vector_type(2)));
typedef float v8f __attribute__((ext_vector_type(8)));

// -------- CDNA5 fp32 WMMA: D(16x16) = A(16x4) * B(4x16) + C --------
__device__ __forceinline__ v8f wmma4(v2f a, v2f b, v8f c) {
  return __builtin_amdgcn_wmma_f32_16x16x4_f32(
      /*neg_a=*/false, a, /*neg_b=*/false, b,
      /*c_mod=*/(short)0, c, /*reuse_a=*/false, /*reuse_b=*/false);
}

__device__ __forceinline__ float gelu_f(float x) {
  const float c = 0.7978845608028654f;  // sqrt(2/pi)
  float u = c * (x + 0.044715f * x * x * x);
  return 0.5f * x * (1.0f + tanhf(u));
}
__device__ __forceinline__ float gelu_grad(float x) {
  const float c = 0.7978845608028654f;
  float x2 = x * x;
  float u = c * (x + 0.044715f * x2 * x);
  float t = tanhf(u);
  return 0.5f * (1.0f + t) + 0.5f * x * (1.0f - t * t) * c * (1.0f + 0.134145f * x2);
}
__device__ __forceinline__ float wave_sum32(float v) {
#pragma unroll
  for (int off = 16; off > 0; off >>= 1) v += __shfl_xor(v, off, 32);
  return v;
}

// ===================================================================
// Kernel 1/4: fp32 WMMA GEMM  C(MxN) = A(MxK) * B(KxN), optional gelu.
// Block = 8 waves (256 thr): one 16-row strip, 8 adjacent 16-col tiles,
// A tile staged once in LDS (coalesced), B streamed (coalesced).
// ===================================================================
__global__ void __launch_bounds__(256)
gemm_wmma_f32(const float* __restrict__ A, const float* __restrict__ Bm,
              float* __restrict__ C, int M, int N, int K, int act) {
  __shared__ float As[16 * 64];
  const int tid   = threadIdx.x;
  const int wave  = tid >> 5;
  const int lane  = tid & 31;
  const int mbase = blockIdx.y * 16;
  const int tn    = blockIdx.x * 8 + wave;
  const int col   = tn * 16 + (lane & 15);
  const int kh    = (lane >> 4) << 1;     // lanes 0-15 -> K+0,1 ; 16-31 -> K+2,3
  const int arow  = lane & 15;
  v8f acc = {0.f, 0.f, 0.f, 0.f, 0.f, 0.f, 0.f, 0.f};
  for (int kc = 0; kc < K; kc += 64) {
    __syncthreads();
    for (int i = tid; i < 16 * 64; i += 256)
      As[i] = A[(size_t)(mbase + (i >> 6)) * K + kc + (i & 63)];
    if (kc + 64 < K)  // hint next A chunk into cache (global_prefetch)
      __builtin_prefetch(&A[(size_t)(mbase + (tid >> 4)) * K + kc + 64], 0, 1);
    __syncthreads();
#pragma unroll
    for (int k0 = 0; k0 < 64; k0 += 4) {
      v2f a, b;
      a.x = As[arow * 64 + k0 + kh];
      a.y = As[arow * 64 + k0 + kh + 1];
      b.x = Bm[(size_t)(kc + k0 + kh) * N + col];
      b.y = Bm[(size_t)(kc + k0 + kh + 1) * N + col];
      acc = wmma4(a, b, acc);
    }
  }
  const int r0 = mbase + ((lane >= 16) ? 8 : 0);
#pragma unroll
  for (int i = 0; i < 8; ++i) {
    float v = acc[i];
    if (act) v = gelu_f(v);
    C[(size_t)(r0 + i) * N + col] = v;
  }
}

// ===================================================================
// Small-GEMM helpers for the TTT scan (operands live in LDS).
// ===================================================================
#define EPI_BIAS      0  // C = acc + bias
#define EPI_GELU_PAIR 1  // C = acc+bias ; C2 = gelu(C)
#define EPI_DPRED     2  // C = GSCALE*(acc + bias - aux[.])
#define EPI_DGELU     3  // C = acc * gelu_grad(aux[.])

// C8(8x64) = A8(8x64, zero-padded to 16 rows) * B(64x64) [+bias],
// B optionally transposed, fused epilogue. Waves 0..3 own one tile each.
__device__ __forceinline__ void
gemm16_rowA(const float* A8, const float* Bm, const float* bias,
            float* C8, float* C2, const float* aux,
            int transB, int epi, int wave, int lane) {
  if (wave >= 4) return;
  const int col  = wave * 16 + (lane & 15);
  const int mrow = lane & 15;
  const int kh   = (lane >> 4) << 1;
  const bool live = (mrow < 8);
  v8f acc = {0.f, 0.f, 0.f, 0.f, 0.f, 0.f, 0.f, 0.f};
#pragma unroll
  for (int k0 = 0; k0 < 64; k0 += 4) {
    v2f a, b;
    a.x = live ? A8[mrow * 64 + k0 + kh] : 0.f;
    a.y = live ? A8[mrow * 64 + k0 + kh + 1] : 0.f;
    if (!transB) {
      b.x = Bm[(k0 + kh) * 64 + col];
      b.y = Bm[(k0 + kh + 1) * 64 + col];
    } else {
      b.x = Bm[col * 64 + k0 + kh];
      b.y = Bm[col * 64 + k0 + kh + 1];
    }
    acc = wmma4(a, b, acc);
  }
  if (lane < 16) {
    const float bv = bias ? bias[col] : 0.f;
#pragma unroll
    for (int i = 0; i < 8; ++i) {
      const int o = i * 64 + col;
      float v = acc[i] + bv;
      if (epi == EPI_GELU_PAIR)      { C8[o] = v; C2[o] = gelu_f(v); }
      else if (epi == EPI_DPRED)     { C8[o] = GSCALE * (v - aux[o]); }
      else if (epi == EPI_DGELU)     { C8[o] = v * gelu_grad(aux[o]); }
      else                           { C8[o] = v; }
    }
  }
}

// W(64x64) -= lr * X8^T(64x8) * Y8(8x64). All 8 waves, 2 tiles each,
// update applied in place straight from WMMA accumulators.
__device__ __forceinline__ void
gemm_outer_update(const float* X8, const float* Y8, float* Wt, float lr,
                  int wave, int lane) {
  const int kh = (lane >> 4) << 1;
#pragma unroll
  for (int ti = 0; ti < 2; ++ti) {
    const int t  = wave * 2 + ti;
    const int tm = t >> 2, tn = t & 3;
    const int m = tm * 16 + (lane & 15);
    const int n = tn * 16 + (lane & 15);
    v8f acc = {0.f, 0.f, 0.f, 0.f, 0.f, 0.f, 0.f, 0.f};
#pragma unroll
    for (int k0 = 0; k0 < 8; k0 += 4) {
      v2f a, b;
      a.x = X8[(k0 + kh) * 64 + m];
      a.y = X8[(k0 + kh + 1) * 64 + m];
      b.x = Y8[(k0 + kh) * 64 + n];
      b.y = Y8[(k0 + kh + 1) * 64 + n];
      acc = wmma4(a, b, acc);
    }
    const int rbase = tm * 16 + ((lane >= 16) ? 8 : 0);
    const int cc    = tn * 16 + (lane & 15);
#pragma unroll
    for (int i = 0; i < 8; ++i) Wt[(rbase + i) * 64 + cc] -= lr * acc[i];
  }
}

// Forward through the fast-weight MLP + its LayerNorm (per head, in LDS).
// gelu fused into the first GEMM's epilogue.
__device__ __forceinline__ void
fwd_pass(const float* ktb, float* av, float* hg, float* ob, float* xh, float* zb,
         const float* W1, const float* W2, const float* b1v, const float* b2v,
         const float* lgv, const float* lbv, float* muv, float* isv,
         int wave, int lane, int tid) {
  gemm16_rowA(ktb, W1, b1v, av, hg, nullptr, 0, EPI_GELU_PAIR, wave, lane);
  __syncthreads();
  gemm16_rowA(hg, W2, b2v, ob, nullptr, nullptr, 0, EPI_BIAS, wave, lane);
  __syncthreads();
  {  // wave w reduces batch row w (64 elems, 2/lane)
    float e0 = ob[wave * 64 + lane], e1 = ob[wave * 64 + lane + 32];
    float s = wave_sum32(e0 + e1);
    float q = wave_sum32(e0 * e0 + e1 * e1);
    if (lane == 0) {
      float m = s * (1.f / 64.f);
      muv[wave] = m;
      isv[wave] = rsqrtf(q * (1.f / 64.f) - m * m + LN_EPS);
    }
  }
  __syncthreads();
  for (int idx = tid; idx < 512; idx += 256) {
    int b = idx >> 6, d = idx & 63;
    float xv = (ob[idx] - muv[b]) * isv[b];
    xh[idx] = xv;
    zb[idx] = xv * lgv[d] + lbv[d];
  }
  __syncthreads();
}

// ===================================================================
// Kernel 2: the sequential TTT scan. One block per head (16 blocks,
// 256 threads = 8 waves). Fast weights + loss_w resident in LDS.
// Next-token k/v is prefetched into registers across the grad steps.
// ===================================================================
#define SCAN_SMEM_FLOATS (3 * 4096 + 10 * 512 + 7 * 64 + 32)

__global__ void __launch_bounds__(256)
ttt_scan_kernel(const float* __restrict__ kg, const float* __restrict__ vg,
                const float* __restrict__ qg,
                const float* __restrict__ fw1, const float* __restrict__ fb1,
                const float* __restrict__ fw2, const float* __restrict__ fb2,
                const float* __restrict__ flg, const float* __restrict__ flb,
                const float* __restrict__ lwg, const float* __restrict__ lbg,
                const float* __restrict__ ttg, const float* __restrict__ ttb,
                float* __restrict__ outg) {
  extern __shared__ float smem[];
  float* W1  = smem;            // 64x64 (D,F)
  float* W2  = W1 + 4096;       // 64x64 (F,D)
  float* LW  = W2 + 4096;       // 64x64 loss_w (D,E)
  float* ktb = LW + 4096;       // 8x64 each below
  float* tgt = ktb + 512;
  float* av  = tgt + 512;
  float* hg  = av + 512;
  float* ob  = hg + 512;
  float* xh  = ob + 512;
  float* zb  = xh + 512;
  float* dzb = zb + 512;
  float* dob = dzb + 512;
  float* dab = dob + 512;       // temp: d_pred / d_a
  float* b1v = dab + 512;       // 64 each below
  float* b2v = b1v + 64;
  float* lgv = b2v + 64;
  float* lbv = lgv + 64;
  float* tgv = lbv + 64;
  float* tbv = tgv + 64;
  float* lbz = tbv + 64;
  float* muv = lbz + 64;        // 8 each below
  float* isv = muv + 8;
  float* m1v = isv + 8;
  float* m2v = m1v + 8;

  const int h    = blockIdx.x;
  const int tid  = threadIdx.x;
  const int wave = tid >> 5;
  const int lane = tid & 31;

  // per-thread global offsets for the two k/v elements this thread owns
  const int b0 = tid >> 6,          d0 = tid & 63;
  const int b1i = (tid + 256) >> 6, d1i = (tid + 256) & 63;
  const size_t g0base = (size_t)(b0 * SS) * HIDC + h * 64 + d0;
  const size_t g1base = (size_t)(b1i * SS) * HIDC + h * 64 + d1i;

  // ---- init fast weights for this head ----
  for (int i = tid; i < 4096; i += 256) {
    W1[i] = fw1[(size_t)h * 4096 + i];   // (H,D,F) -> [d*64+f]
    W2[i] = fw2[(size_t)h * 4096 + i];   // (H,F,D) -> [f*64+d]
    LW[i] = lwg[i];
  }
  if (tid < 64) {
    b1v[tid] = fb1[h * 64 + tid];
    b2v[tid] = fb2[h * 64 + tid];
    lgv[tid] = flg[h * 64 + tid];
    lbv[tid] = flb[h * 64 + tid];
    tgv[tid] = ttg[h * 64 + tid];
    tbv[tid] = ttb[h * 64 + tid];
    lbz[tid] = lbg[tid];
  }

  // prefetch token 0
  float kp0 = kg[g0base], vp0 = vg[g0base];
  float kp1 = kg[g1base], vp1 = vg[g1base];
  __syncthreads();

  for (int t = 0; t < SS; ++t) {
    // ---- commit current token's k/target to LDS ----
    ktb[tid]       = kp0;  tgt[tid]       = vp0 - kp0;
    ktb[tid + 256] = kp1;  tgt[tid + 256] = vp1 - kp1;
    __syncthreads();
    // ---- issue next token's loads; they stay in flight across all steps ----
    if (t + 1 < SS) {
      kp0 = kg[g0base + (t + 1) * HIDC];  vp0 = vg[g0base + (t + 1) * HIDC];
      kp1 = kg[g1base + (t + 1) * HIDC];  vp1 = vg[g1base + (t + 1) * HIDC];
    }

    for (int step = 0; step < 4; ++step) {
      // forward: av, hg (fused gelu), ob, xh, zb (+mu/invsigma)
      fwd_pass(ktb, av, hg, ob, xh, zb, W1, W2, b1v, b2v, lgv, lbv,
               muv, isv, wave, lane, tid);
      // d_pred = 2*((zb@LW + loss_b) - target)/(B*H*D), fused epilogue
      gemm16_rowA(zb, LW, lbz, dab, nullptr, tgt, 0, EPI_DPRED, wave, lane);
      __syncthreads();
      // d_z = d_pred @ LW^T
      gemm16_rowA(dab, LW, nullptr, dzb, nullptr, nullptr, 1, EPI_BIAS, wave, lane);
      __syncthreads();
      // LN backward row means: m1 = mean(dxhat), m2 = mean(dxhat*xhat)
      {
        float dd0 = dzb[wave * 64 + lane] * lgv[lane];
        float dd1 = dzb[wave * 64 + lane + 32] * lgv[lane + 32];
        float x0 = xh[wave * 64 + lane], x1 = xh[wave * 64 + lane + 32];
        float s1 = wave_sum32(dd0 + dd1);
        float s2 = wave_sum32(dd0 * x0 + dd1 * x1);
        if (lane == 0) { m1v[wave] = s1 * (1.f / 64.f); m2v[wave] = s2 * (1.f / 64.f); }
      }
      __syncthreads();
      for (int idx = tid; idx < 512; idx += 256) {
        int b = idx >> 6, d = idx & 63;
        float dxh = dzb[idx] * lgv[d];
        dob[idx] = isv[b] * (dxh - m1v[b] - xh[idx] * m2v[b]);
      }
      __syncthreads();
      // LN param grads (read dzb/xh, write lgv/lbv; lgv reads are done)
      if (tid < 64) {
        float s = 0.f;
        for (int b = 0; b < 8; ++b) s += dzb[b * 64 + tid] * xh[b * 64 + tid];
        lgv[tid] -= LR * s;
      } else if (tid < 128) {
        int d = tid - 64; float s = 0.f;
        for (int b = 0; b < 8; ++b) s += dzb[b * 64 + d];
        lbv[d] -= LR * s;
      }
      // d_a = (dob @ W2^T) * gelu'(a), fused epilogue (uses OLD W2)
      gemm16_rowA(dob, W2, nullptr, dab, nullptr, av, 1, EPI_DGELU, wave, lane);
      __syncthreads();
      // all gradients ready: apply every weight update in one region
      gemm_outer_update(hg, dob, W2, LR, wave, lane);   // W2 -= lr*hg^T dob
      gemm_outer_update(ktb, dab, W1, LR, wave, lane);  // W1 -= lr*kt^T d_a
      if (tid < 64) {
        float s = 0.f;
        for (int b = 0; b < 8; ++b) s += dob[b * 64 + tid];
        b2v[tid] -= LR * s;
      } else if (tid < 128) {
        int d = tid - 64; float s = 0.f;
        for (int b = 0; b < 8; ++b) s += dab[b * 64 + d];
        b1v[d] -= LR * s;
      }
      __syncthreads();
    }

    // final apply with updated weights
    fwd_pass(ktb, av, hg, ob, xh, zb, W1, W2, b1v, b2v, lgv, lbv,
             muv, isv, wave, lane, tid);
    // second LN (ttt_g/ttt_b) over zb, out = q + z2
    {
      float e0 = zb[wave * 64 + lane], e1 = zb[wave * 64 + lane + 32];
      float s = wave_sum32(e0 + e1);
      float q = wave_sum32(e0 * e0 + e1 * e1);
      if (lane == 0) {
        float m = s * (1.f / 64.f);
        muv[wave] = m;
        isv[wave] = rsqrtf(q * (1.f / 64.f) - m * m + LN_EPS);
      }
    }
    __syncthreads();
    for (int idx = tid; idx < 512; idx += 256) {
      int b = idx >> 6, d = idx & 63;
      float z2 = (zb[idx] - muv[b]) * isv[b] * tgv[d] + tbv[d];
      size_t g = ((size_t)(b * SS + t)) * HIDC + h * 64 + d;
      outg[g] = qg[g] + z2;
    }
    __syncthreads();
  }
}

// ===================================================================
// Kernel 3: post-norm over HID fused with the gelu gate.
// gated = gelu(x@wg) * (LN_1024(out)*pn_g + pn_b)
// ===================================================================
__global__ void __launch_bounds__(256)
gate_ln_kernel(const float* __restrict__ outb, const float* __restrict__ gact,
               const float* __restrict__ png, const float* __restrict__ pnb,
               float* __restrict__ gated) {
  __shared__ float r1[256], r2[256];
  __shared__ float mu_s, inv_s;
  const int r = blockIdx.x, tid = threadIdx.x;
  const float* row = outb + (size_t)r * HIDC;
  float s = 0.f, q = 0.f;
  for (int i = tid; i < HIDC; i += 256) { float v = row[i]; s += v; q += v * v; }
  r1[tid] = s; r2[tid] = q;
  __syncthreads();
  for (int off = 128; off > 0; off >>= 1) {
    if (tid < off) { r1[tid] += r1[tid + off]; r2[tid] += r2[tid + off]; }
    __syncthreads();
  }
  if (tid == 0) {
    float m = r1[0] * (1.f / HIDC);
    mu_s = m;
    inv_s = rsqrtf(r2[0] * (1.f / HIDC) - m * m + LN_EPS);
  }
  __syncthreads();
  for (int i = tid; i < HIDC; i += 256) {
    float y = (row[i] - mu_s) * inv_s * png[i] + pnb[i];
    gated[(size_t)r * HIDC + i] = gact[(size_t)r * HIDC + i] * y;
  }
}

// ===================================================================
extern "C" void kernel_launch(void* const* d_in, const int* in_sizes, int n_in,
                              void* d_out, int out_size, void* d_ws, size_t ws_size,
                              hipStream_t stream) {
  (void)in_sizes; (void)n_in; (void)out_size; (void)ws_size;
  const float* x   = (const float*)d_in[0];
  const float* wq  = (const float*)d_in[1];
  const float* wk  = (const float*)d_in[2];
  const float* wv  = (const float*)d_in[3];
  const float* fw1 = (const float*)d_in[4];
  const float* fb1 = (const float*)d_in[5];
  const float* fw2 = (const float*)d_in[6];
  const float* fb2 = (const float*)d_in[7];
  const float* flg = (const float*)d_in[8];
  const float* flb = (const float*)d_in[9];
  const float* lw  = (const float*)d_in[10];
  const float* lb  = (const float*)d_in[11];
  const float* ttg = (const float*)d_in[12];
  const float* ttb = (const float*)d_in[13];
  const float* wo  = (const float*)d_in[14];
  const float* wg  = (const float*)d_in[15];
  const float* png = (const float*)d_in[16];
  const float* pnb = (const float*)d_in[17];

  const int M = BB * SS;                  // 2048
  const size_t PLANE = (size_t)M * HIDC;  // 2M floats = 8MB
  float* qb    = (float*)d_ws;
  float* kb    = qb + PLANE;
  float* vb    = kb + PLANE;
  float* gactb = vb + PLANE;
  float* outb  = gactb + PLANE;           // needs >= 40MB workspace total
  float* gatedb = qb;                     // q is dead after the scan; reuse

  dim3 gblk(256);
  dim3 ggrid(HIDC / 128, M / 16);         // (8, 128): 8 col-tiles per block
  gemm_wmma_f32<<<ggrid, gblk, 0, stream>>>(x, wq, qb,    M, HIDC, HIDC, 0);
  gemm_wmma_f32<<<ggrid, gblk, 0, stream>>>(x, wk, kb,    M, HIDC, HIDC, 0);
  gemm_wmma_f32<<<ggrid, gblk, 0, stream>>>(x, wv, vb,    M, HIDC, HIDC, 0);
  gemm_wmma_f32<<<ggrid, gblk, 0, stream>>>(x, wg, gactb, M, HIDC, HIDC, 1);

  ttt_scan_kernel<<<HH, 256, SCAN_SMEM_FLOATS * sizeof(float), stream>>>(
      kb, vb, qb, fw1, fb1, fw2, fb2, flg, flb, lw, lb, ttg, ttb, outb);

  gate_ln_kernel<<<M, 256, 0, stream>>>(outb, gactb, png, pnb, gatedb);

  gemm_wmma_f32<<<ggrid, gblk, 0, stream>>>(gatedb, wo, (float*)d_out,
                                            M, HIDC, HIDC, 0);
}